// TemporalKplanesEncoding_3298534884030
// MI455X (gfx1250) — compile-verified
//
#include <hip/hip_runtime.h>
#include <hip/hip_bf16.h>

// ---------------------------------------------------------------------------
// TemporalKplanesEncoding (K-planes, reduce='sum') for MI455X / gfx1250.
//
// Planes: 3 x [F=32, H=128, W=256] f32.  Points: 2,097,152 x 4 (x,y,z,t).
// Out:    [N, 32] f32 = sum of 3 bilinear lookups (align_corners, border).
//
// Strategy (memory-bound, WMMA inapplicable — gathers, no shared operand):
//  1) One-time per-launch transpose of each plane to [H, W, F] in d_ws so a
//     texel's 32 features are contiguous (128 B = one wave32 transaction).
//  2) Sample kernel: wave = point, lane = feature. Every corner gather and
//     the output store are fully coalesced 128 B accesses. 4 points/wave for
//     memory-level parallelism.
//  3) Cache control: output (256 MB) and coords (32 MB) use non-temporal
//     hints so the 12 MB of transposed planes stay resident in the 192 MB L2.
// ---------------------------------------------------------------------------

#define FEAT 32
#define PH   128          // H = RES[3]
#define PW   256          // W = RES[comb[0]]
#define PHW  (PH * PW)    // 32768 texels per plane
#define NPTS 2097152
#define PPW  4            // points per wave
#define WPB  8            // waves per block (256 threads)

typedef float v4f __attribute__((ext_vector_type(4)));

// ---------------------------------------------------------------------------
// Transpose [F, H*W] -> [H*W, F] via LDS 32x32 tiles (conflict-free, padded).
// Reads are NT (planes never re-read in original layout); writes are RT so the
// transposed planes become L2-resident for the sample kernel.
// ---------------------------------------------------------------------------
__global__ __launch_bounds__(1024) void kplanes_transpose(
    const float* __restrict__ p0, const float* __restrict__ p1,
    const float* __restrict__ p2, float* __restrict__ dstAll)
{
    __shared__ float tile[32][33];

    const int plane = blockIdx.x >> 10;          // 1024 tiles per plane
    const int hw0   = (blockIdx.x & 1023) << 5;  // 32 texels per tile
    const float* __restrict__ src =
        (plane == 0) ? p0 : (plane == 1) ? p1 : p2;
    float* __restrict__ dst = dstAll + (size_t)plane * (FEAT * PHW);

    const int tx = threadIdx.x & 31;   // texel-in-tile on load, feature on store
    const int ty = threadIdx.x >> 5;   // feature on load, texel-in-tile on store

    // Coalesced NT read: 32 consecutive texels of feature row ty.
    tile[ty][tx] = __builtin_nontemporal_load(&src[(size_t)ty * PHW + hw0 + tx]);
    __syncthreads();
    // Coalesced RT write: features 0..31 of texel hw0+ty are contiguous 128 B.
    dst[(size_t)(hw0 + ty) * FEAT + tx] = tile[tx][ty];
}

// ---------------------------------------------------------------------------
// Sample kernel. TR=true: planes in [H,W,F] layout (ws); TR=false: original
// [F,H,W] layout fallback (strided gathers) when ws is too small.
// ---------------------------------------------------------------------------
template <bool TR>
__global__ __launch_bounds__(256) void kplanes_sample(
    const float* __restrict__ inp,
    const float* __restrict__ q0, const float* __restrict__ q1,
    const float* __restrict__ q2, float* __restrict__ out)
{
    const int lane  = threadIdx.x & 31;   // feature index
    const int wave  = threadIdx.x >> 5;
    const int pbase = (blockIdx.x * WPB + wave) * PPW;

#pragma unroll
    for (int k = 0; k < PPW; ++k) {
        const int p = pbase + k;

        // All 32 lanes read the same 16 B (hardware broadcast); NT hint —
        // coords are streamed exactly once.
        const v4f c =
            __builtin_nontemporal_load((const v4f*)inp + p);

        // y (time axis) is shared by all three planes: H = 128.
        float y  = fminf(fmaxf((c[3] + 1.0f) * (0.5f * (PH - 1)), 0.0f),
                         (float)(PH - 1));
        float y0f = floorf(y);
        float wy  = y - y0f;
        int   y0  = (int)y0f;
        int   y1  = min(y0 + 1, PH - 1);

        const float cxs[3] = {c[0], c[1], c[2]};
        const float* __restrict__ planes[3] = {q0, q1, q2};

        float acc = 0.0f;
#pragma unroll
        for (int pl = 0; pl < 3; ++pl) {
            float x  = fminf(fmaxf((cxs[pl] + 1.0f) * (0.5f * (PW - 1)), 0.0f),
                             (float)(PW - 1));
            float x0f = floorf(x);
            float wx  = x - x0f;
            int   x0  = (int)x0f;
            int   x1  = min(x0 + 1, PW - 1);

            const float* __restrict__ P = planes[pl];
            float v00, v01, v10, v11;
            if (TR) {
                // [H,W,F]: lane-consecutive addresses -> one 128 B L2 txn each.
                v00 = P[(size_t)(y0 * PW + x0) * FEAT + lane];
                v01 = P[(size_t)(y0 * PW + x1) * FEAT + lane];
                v10 = P[(size_t)(y1 * PW + x0) * FEAT + lane];
                v11 = P[(size_t)(y1 * PW + x1) * FEAT + lane];
            } else {
                // [F,H,W] fallback: 128 KB stride between lanes.
                const float* __restrict__ Pf = P + (size_t)lane * PHW;
                v00 = Pf[y0 * PW + x0];
                v01 = Pf[y0 * PW + x1];
                v10 = Pf[y1 * PW + x0];
                v11 = Pf[y1 * PW + x1];
            }

            const float w00 = (1.0f - wx) * (1.0f - wy);
            const float w01 = wx * (1.0f - wy);
            const float w10 = (1.0f - wx) * wy;
            const float w11 = wx * wy;
            acc += v00 * w00 + v01 * w01 + v10 * w10 + v11 * w11;
        }

        // 256 MB output stream: NT store keeps planes resident in L2.
        __builtin_nontemporal_store(acc, out + (size_t)p * FEAT + lane);
    }
}

extern "C" void kernel_launch(void* const* d_in, const int* in_sizes, int n_in,
                              void* d_out, int out_size, void* d_ws,
                              size_t ws_size, hipStream_t stream)
{
    const float* inp = (const float*)d_in[0];
    const float* p0  = (const float*)d_in[1];
    const float* p1  = (const float*)d_in[2];
    const float* p2  = (const float*)d_in[3];
    float*       out = (float*)d_out;

    const size_t planeElems = (size_t)FEAT * PHW;          // 1,048,576
    const size_t needBytes  = 3 * planeElems * sizeof(float); // 12 MB

    const int sampleBlocks = NPTS / (WPB * PPW);           // 65536

    if (ws_size >= needBytes) {
        float* t = (float*)d_ws;
        // 3 planes x 1024 tiles, 1024 threads each.
        kplanes_transpose<<<3 * (PHW / 32), 1024, 0, stream>>>(p0, p1, p2, t);
        kplanes_sample<true><<<sampleBlocks, 256, 0, stream>>>(
            inp, t, t + planeElems, t + 2 * planeElems, out);
    } else {
        kplanes_sample<false><<<sampleBlocks, 256, 0, stream>>>(
            inp, p0, p1, p2, out);
    }
}